// UpBlock_53901839564868
// MI455X (gfx1250) — compile-verified
//
#include <hip/hip_runtime.h>

typedef float v2f __attribute__((ext_vector_type(2)));
typedef float v8f __attribute__((ext_vector_type(8)));

namespace {
constexpr int Bb   = 4;
constexpr int Np   = 8192;    // dst points
constexpr int Mp   = 2048;    // src points
constexpr int Cx   = 256;
constexpr int Cs   = 512;
constexpr int Cin  = Cx + Cs; // 768
constexpr int Cout = 256;
constexpr int COLS = Bb * Np; // 32768
}

// ---------------------------------------------------------------------------
// Kernel 1: 3-NN over src points (LDS-resident), inverse-d^2 weights.
// grid (Np/256, Bb), block 256
// ---------------------------------------------------------------------------
__global__ void knn_kernel(const float* __restrict__ xyz,
                           const float* __restrict__ sub_xyz,
                           int* __restrict__ idx, float* __restrict__ wgt) {
    __shared__ float sx[Mp], sy[Mp], sz[Mp];
    const int b   = blockIdx.y;
    const int tid = threadIdx.x;
    const int n   = blockIdx.x * 256 + tid;

    const float* sb = sub_xyz + (size_t)b * 3 * Mp;
    for (int j = tid; j < Mp; j += 256) {
        sx[j] = sb[j];
        sy[j] = sb[Mp + j];
        sz[j] = sb[2 * Mp + j];
    }
    __syncthreads();

    const float px = xyz[((size_t)b * 3 + 0) * Np + n];
    const float py = xyz[((size_t)b * 3 + 1) * Np + n];
    const float pz = xyz[((size_t)b * 3 + 2) * Np + n];

    float d0 = 3.4e38f, d1 = 3.4e38f, d2 = 3.4e38f;
    int   i0 = 0, i1 = 0, i2 = 0;
    for (int j = 0; j < Mp; ++j) {
        const float dx = px - sx[j], dy = py - sy[j], dz = pz - sz[j];
        const float d  = dx * dx + dy * dy + dz * dz;
        if (d < d2) {
            if (d < d1) {
                if (d < d0) { d2 = d1; i2 = i1; d1 = d0; i1 = i0; d0 = d; i0 = j; }
                else        { d2 = d1; i2 = i1; d1 = d;  i1 = j; }
            } else          { d2 = d;  i2 = j; }
        }
    }
    float w0 = 1.0f / (d0 + 1e-8f);
    float w1 = 1.0f / (d1 + 1e-8f);
    float w2 = 1.0f / (d2 + 1e-8f);
    const float inv = 1.0f / (w0 + w1 + w2);
    const size_t base = ((size_t)b * Np + n) * 3;
    idx[base + 0] = i0; idx[base + 1] = i1; idx[base + 2] = i2;
    wgt[base + 0] = w0 * inv; wgt[base + 1] = w1 * inv; wgt[base + 2] = w2 * inv;
}

// ---------------------------------------------------------------------------
// Kernel 2: feature gather+lerp into GEMM-ready layout lerp[c][b*Np+n].
// grid (Np/256, Cs, Bb), block 256
// ---------------------------------------------------------------------------
__global__ void gather_kernel(const float* __restrict__ sub_x,
                              const int* __restrict__ idx,
                              const float* __restrict__ wgt,
                              float* __restrict__ lerp) {
    const int n = blockIdx.x * 256 + threadIdx.x;
    const int c = blockIdx.y;
    const int b = blockIdx.z;
    const size_t base = ((size_t)b * Np + n) * 3;
    const int   i0 = idx[base], i1 = idx[base + 1], i2 = idx[base + 2];
    const float w0 = wgt[base], w1 = wgt[base + 1], w2 = wgt[base + 2];
    const float* row = sub_x + ((size_t)b * Cs + c) * Mp;
    const float v = w0 * row[i0] + w1 * row[i1] + w2 * row[i2];
    lerp[(size_t)c * COLS + (size_t)b * Np + n] = v;
}

// ---------------------------------------------------------------------------
// WMMA f32 GEMM: out[m][col] = sum_k W[m][k] * In[k][col]
// mode 0: In = concat(x[b,c,n] for k<Cx, lerp[k-Cx][col]) , K = Cin
// mode 1: In = relu(a[k]*h1raw[k][col] + bcoef[k])         , K = Cout
//
// block = 256 threads (8 wave32s, 2(M) x 4(N) wave grid).
// Block tile 64(M) x 128(N); each wave computes 32x32 via 4 v8f accumulators
// -> per inner step: 2 A-frags + 2 B-frags feed 4 v_wmma_f32_16x16x4_f32.
// K staged through LDS in chunks of 32 with B128 loads/stores.
// grid (COLS/128, M/64)
// ---------------------------------------------------------------------------
__global__ void gemm_kernel(const float* __restrict__ W,
                            const float* __restrict__ x,
                            const float* __restrict__ srcB,
                            const float* __restrict__ acoef,
                            const float* __restrict__ bcoef,
                            float* __restrict__ out,
                            int K, int mode) {
    __shared__ float As[64][36];   // [m][k], 16B-aligned rows, conflict-free
    __shared__ float Bs[32][132];  // [k][n], 16B-aligned rows, conflict-free

    const int tid  = threadIdx.x;
    const int wave = tid >> 5;
    const int lane = tid & 31;
    const int lm   = lane & 15;
    const int lh   = lane >> 4;
    const int wm   = (wave & 1) * 32;   // wave M offset in 64-row block tile
    const int wn   = (wave >> 1) * 32;  // wave N offset in 128-col block tile

    const int col0 = blockIdx.x * 128;
    const int m0   = blockIdx.y * 64;

    v8f acc00 = {}, acc01 = {}, acc10 = {}, acc11 = {};

    for (int kc = 0; kc < K; kc += 32) {
        // stage A: 64x32 weights as float4 (512 xB128 / 256 threads = 2 each)
        #pragma unroll
        for (int i = 0; i < 2; ++i) {
            const int e  = tid + i * 256;
            const int m  = e >> 3;         // 8 float4 per 32-float row
            const int k4 = (e & 7) * 4;
            const float4 v =
                *(const float4*)(W + (size_t)(m0 + m) * K + (kc + k4));
            *(float4*)(&As[m][k4]) = v;
        }
        // stage B: 32x128 activations as float4 (1024 xB128 / 256 = 4 each)
        #pragma unroll
        for (int i = 0; i < 4; ++i) {
            const int e  = tid + i * 256;
            const int k  = e >> 5;         // 32 float4 per 128-float row
            const int n4 = (e & 31) * 4;
            const int gk  = kc + k;
            const int col = col0 + n4;
            float4 v;
            if (mode == 0) {
                if (gk < Cx) {
                    const int b  = col >> 13;        // col / Np
                    const int nn = col & (Np - 1);
                    v = *(const float4*)(x + ((size_t)b * Cx + gk) * Np + nn);
                } else {
                    v = *(const float4*)(srcB + (size_t)(gk - Cx) * COLS + col);
                }
            } else {
                v = *(const float4*)(srcB + (size_t)gk * COLS + col);
                const float ac = acoef[gk], bc = bcoef[gk];
                v.x = fmaxf(fmaf(ac, v.x, bc), 0.0f);   // BN1 + ReLU fused
                v.y = fmaxf(fmaf(ac, v.y, bc), 0.0f);
                v.z = fmaxf(fmaf(ac, v.z, bc), 0.0f);
                v.w = fmaxf(fmaf(ac, v.w, bc), 0.0f);
            }
            *(float4*)(&Bs[k][n4]) = v;
        }
        __syncthreads();

        #pragma unroll
        for (int ko = 0; ko < 32; ko += 4) {
            v2f a0, a1, b0, b1;
            a0.x = As[wm + lm][ko + lh * 2 + 0];
            a0.y = As[wm + lm][ko + lh * 2 + 1];
            a1.x = As[wm + 16 + lm][ko + lh * 2 + 0];
            a1.y = As[wm + 16 + lm][ko + lh * 2 + 1];
            b0.x = Bs[ko + lh * 2 + 0][wn + lm];
            b0.y = Bs[ko + lh * 2 + 1][wn + lm];
            b1.x = Bs[ko + lh * 2 + 0][wn + 16 + lm];
            b1.y = Bs[ko + lh * 2 + 1][wn + 16 + lm];
            acc00 = __builtin_amdgcn_wmma_f32_16x16x4_f32(
                false, a0, false, b0, (short)0, acc00, false, false);
            acc01 = __builtin_amdgcn_wmma_f32_16x16x4_f32(
                false, a0, false, b1, (short)0, acc01, false, false);
            acc10 = __builtin_amdgcn_wmma_f32_16x16x4_f32(
                false, a1, false, b0, (short)0, acc10, false, false);
            acc11 = __builtin_amdgcn_wmma_f32_16x16x4_f32(
                false, a1, false, b1, (short)0, acc11, false, false);
        }
        __syncthreads();
    }

    const int colA = col0 + wn + lm;
    const int colB = colA + 16;
    #pragma unroll
    for (int r = 0; r < 8; ++r) {
        const int mA = m0 + wm + r + lh * 8;
        const int mB = mA + 16;
        out[(size_t)mA * COLS + colA] = acc00[r];
        out[(size_t)mA * COLS + colB] = acc01[r];
        out[(size_t)mB * COLS + colA] = acc10[r];
        out[(size_t)mB * COLS + colB] = acc11[r];
    }
}

// ---------------------------------------------------------------------------
// Per-channel BN training stats -> folded affine (a, b). grid (Cout), block 256
// ---------------------------------------------------------------------------
__global__ void stats_kernel(const float* __restrict__ h,
                             const float* __restrict__ gamma,
                             const float* __restrict__ beta,
                             float* __restrict__ a, float* __restrict__ b) {
    __shared__ float rs[256], rs2[256];
    const int c   = blockIdx.x;
    const int tid = threadIdx.x;
    const float* row = h + (size_t)c * COLS;
    float s = 0.0f, s2 = 0.0f;
    for (int i = tid; i < COLS; i += 256) {
        const float v = row[i];
        s += v; s2 += v * v;
    }
    rs[tid] = s; rs2[tid] = s2;
    __syncthreads();
    for (int off = 128; off > 0; off >>= 1) {
        if (tid < off) { rs[tid] += rs[tid + off]; rs2[tid] += rs2[tid + off]; }
        __syncthreads();
    }
    if (tid == 0) {
        const float mean = rs[0] * (1.0f / COLS);
        const float var  = rs2[0] * (1.0f / COLS) - mean * mean; // biased
        const float sc   = gamma[c] * rsqrtf(var + 1e-5f);
        a[c] = sc;
        b[c] = beta[c] - mean * sc;
    }
}

// ---------------------------------------------------------------------------
// BN2 + ReLU + layout permute [o][b*Np+n] -> [b][o][n]. grid (Np/256, Cout, Bb)
// ---------------------------------------------------------------------------
__global__ void finalize_kernel(const float* __restrict__ h2,
                                const float* __restrict__ a2,
                                const float* __restrict__ b2,
                                float* __restrict__ out) {
    const int n = blockIdx.x * 256 + threadIdx.x;
    const int o = blockIdx.y;
    const int b = blockIdx.z;
    const float v = h2[(size_t)o * COLS + (size_t)b * Np + n];
    out[((size_t)b * Cout + o) * Np + n] = fmaxf(fmaf(a2[o], v, b2[o]), 0.0f);
}

// ---------------------------------------------------------------------------
extern "C" void kernel_launch(void* const* d_in, const int* in_sizes, int n_in,
                              void* d_out, int out_size, void* d_ws, size_t ws_size,
                              hipStream_t stream) {
    const float* x       = (const float*)d_in[0]; // [4,256,8192]
    const float* xyz     = (const float*)d_in[1]; // [4,3,8192]
    const float* sub_x   = (const float*)d_in[2]; // [4,512,2048]
    const float* sub_xyz = (const float*)d_in[3]; // [4,3,2048]
    const float* w1      = (const float*)d_in[4]; // [256,768]
    const float* g1      = (const float*)d_in[5];
    const float* be1     = (const float*)d_in[6];
    const float* w2      = (const float*)d_in[7]; // [256,256]
    const float* g2      = (const float*)d_in[8];
    const float* be2     = (const float*)d_in[9];
    float* out = (float*)d_out;

    char* ws = (char*)d_ws;
    size_t off = 0;
    auto alloc = [&](size_t bytes) {
        void* p = ws + off;
        off += (bytes + 255) & ~(size_t)255;
        return p;
    };
    int*   idx   = (int*)  alloc((size_t)Bb * Np * 3 * sizeof(int));
    float* wgt   = (float*)alloc((size_t)Bb * Np * 3 * sizeof(float));
    float* lerp  = (float*)alloc((size_t)Cs * COLS * sizeof(float));
    float* h1raw = (float*)alloc((size_t)Cout * COLS * sizeof(float));
    float* h2raw = (float*)alloc((size_t)Cout * COLS * sizeof(float));
    float* a1    = (float*)alloc(Cout * sizeof(float));
    float* b1c   = (float*)alloc(Cout * sizeof(float));
    float* a2    = (float*)alloc(Cout * sizeof(float));
    float* b2c   = (float*)alloc(Cout * sizeof(float));
    (void)in_sizes; (void)n_in; (void)out_size; (void)ws_size;

    knn_kernel<<<dim3(Np / 256, Bb), 256, 0, stream>>>(xyz, sub_xyz, idx, wgt);

    gather_kernel<<<dim3(Np / 256, Cs, Bb), 256, 0, stream>>>(sub_x, idx, wgt, lerp);

    gemm_kernel<<<dim3(COLS / 128, Cout / 64), 256, 0, stream>>>(
        w1, x, lerp, nullptr, nullptr, h1raw, Cin, 0);

    stats_kernel<<<dim3(Cout), 256, 0, stream>>>(h1raw, g1, be1, a1, b1c);

    gemm_kernel<<<dim3(COLS / 128, Cout / 64), 256, 0, stream>>>(
        w2, nullptr, h1raw, a1, b1c, h2raw, Cout, 1);

    stats_kernel<<<dim3(Cout), 256, 0, stream>>>(h2raw, g2, be2, a2, b2c);

    finalize_kernel<<<dim3(Np / 256, Cout, Bb), 256, 0, stream>>>(h2raw, a2, b2c, out);
}